// mowLSTM__3624952398091
// MI455X (gfx1250) — compile-verified
//
#include <hip/hip_runtime.h>
#include <hip/hip_bf16.h>
#include <math.h>

// Problem constants (from reference): T,B,I,H,E
#define T_ 2048
#define B_ 32
#define I_ 256
#define H_ 256
#define E_ 4
#define G_ 1024   // 4*H

typedef __attribute__((ext_vector_type(2))) float v2f;
typedef __attribute__((ext_vector_type(8))) float v8f;

// ---------------------------------------------------------------------------
// Kernel 1: collapse experts into per-batch effective weights, stored K-major:
//   WxT[b][i][o] = sum_e coef[b,e] * Wx[e,o,i]
//   WhT[b][h][o] = sum_e coef[b,e] * Wh[e,o,h]
// ---------------------------------------------------------------------------
__global__ void mow_eff_weights(const float* __restrict__ coef,
                                const float* __restrict__ Wx,
                                const float* __restrict__ Wh,
                                float* __restrict__ WxT,
                                float* __restrict__ WhT) {
    const size_t N1 = (size_t)B_ * I_ * G_;
    size_t idx = (size_t)blockIdx.x * blockDim.x + threadIdx.x;
    const float* __restrict__ src;
    float* __restrict__ dst;
    size_t j;
    if (idx < N1) { src = Wx; dst = WxT; j = idx; }
    else          { src = Wh; dst = WhT; j = idx - N1; }
    int b = (int)(j / ((size_t)I_ * G_));
    size_t rem = j % ((size_t)I_ * G_);
    int i = (int)(rem / G_);
    int o = (int)(rem % G_);
    float s = 0.f;
#pragma unroll
    for (int e = 0; e < E_; ++e)
        s += coef[b * E_ + e] * src[((size_t)e * G_ + o) * I_ + i];
    dst[j] = s;
}

// ---------------------------------------------------------------------------
// Kernel 2: input projection, per-batch GEMM with fp32 WMMA.
//   xg[t][b][o] = sum_i x[t,b,i] * WxT[b][i][o]
// Each wave computes a 16(M=t) x 64(N=o) strip of C for one batch.
// A (16x4 f32): lane L(0-15): v0=A[L][k0],v1=A[L][k0+1]; lanes 16-31: k0+2,k0+3
// B (4x16 f32): v0: lanes0-15=B[k0][n+L], lanes16-31=B[k0+2][n+L-16]; v1: k0+1/k0+3
// ---------------------------------------------------------------------------
__global__ void mow_xproj_gemm(const float* __restrict__ x,
                               const float* __restrict__ WxT,
                               float* __restrict__ xg) {
    const int lane = threadIdx.x & 31;
    const int wid  = threadIdx.x >> 5;
    const int lrow = lane & 15;
    const int half = lane >> 4;

    // 2048 waves per batch: 128 m-tiles x 16 n-strips
    int wave = blockIdx.x * (blockDim.x >> 5) + wid;
    int b      = wave >> 11;
    int rem    = wave & 2047;
    int t0     = (rem >> 4) << 4;   // m-tile * 16
    int n0     = (rem & 15) << 6;   // n-strip * 64

    const float* __restrict__ xrow = x + ((size_t)(t0 + lrow) * B_ + b) * I_ + 2 * half;
    const float* __restrict__ wB   = WxT + (size_t)b * I_ * G_ + n0 + lrow;

    v8f acc0 = {}, acc1 = {}, acc2 = {}, acc3 = {};

#pragma unroll 4
    for (int k0 = 0; k0 < I_; k0 += 4) {
        v2f a = *(const v2f*)(xrow + k0);                 // one b64 load
        const float* wk = wB + (size_t)(k0 + 2 * half) * G_;
        v2f b0, b1, b2, b3;
        b0.x = wk[0];  b0.y = wk[G_ + 0];
        b1.x = wk[16]; b1.y = wk[G_ + 16];
        b2.x = wk[32]; b2.y = wk[G_ + 32];
        b3.x = wk[48]; b3.y = wk[G_ + 48];
        acc0 = __builtin_amdgcn_wmma_f32_16x16x4_f32(false, a, false, b0, (short)0, acc0, false, false);
        acc1 = __builtin_amdgcn_wmma_f32_16x16x4_f32(false, a, false, b1, (short)0, acc1, false, false);
        acc2 = __builtin_amdgcn_wmma_f32_16x16x4_f32(false, a, false, b2, (short)0, acc2, false, false);
        acc3 = __builtin_amdgcn_wmma_f32_16x16x4_f32(false, a, false, b3, (short)0, acc3, false, false);
    }

    // C layout: VGPR r, lanes0-15 -> M=r, N=lane; lanes16-31 -> M=8+r, N=lane-16
    const int tbase = t0 + 8 * half;
    const int obase = n0 + lrow;
#pragma unroll
    for (int r = 0; r < 8; ++r) {
        size_t row = ((size_t)(tbase + r) * B_ + b) * G_ + obase;
        xg[row +  0] = acc0[r];
        xg[row + 16] = acc1[r];
        xg[row + 32] = acc2[r];
        xg[row + 48] = acc3[r];
    }
}

// ---------------------------------------------------------------------------
// Kernel 3: sequential LSTM recurrence. One 1024-thread workgroup per batch.
// Thread o owns gate output o; h and gates live in LDS. Weights stream from L2.
// ---------------------------------------------------------------------------
__global__ void mow_recurrence(const float* __restrict__ x,
                               const float* __restrict__ xg,
                               const float* __restrict__ h0,
                               const float* __restrict__ c0,
                               const float* __restrict__ coef,
                               const float* __restrict__ bx,
                               const float* __restrict__ bh,
                               const float* __restrict__ WxT,
                               const float* __restrict__ WhT,
                               float* __restrict__ out,
                               int use_xg) {
    __shared__ float h_lds[H_];
    __shared__ float gates[G_];
    __shared__ float x_lds[I_];

    const int b = blockIdx.x;
    const int o = threadIdx.x;  // 0..1023

    float bias = 0.f;
#pragma unroll
    for (int e = 0; e < E_; ++e)
        bias += coef[b * E_ + e] * (bx[e * G_ + o] + bh[e * G_ + o]);

    float c = 0.f;
    if (o < H_) {
        h_lds[o] = h0[b * H_ + o];
        c = c0[b * H_ + o];
    }
    __syncthreads();

    const float* __restrict__ wh = WhT + (size_t)b * H_ * G_ + o;
    const float* __restrict__ wx = WxT + (size_t)b * I_ * G_ + o;

    for (int t = 0; t < T_; ++t) {
        if (!use_xg) {
            if (o < I_) x_lds[o] = x[((size_t)t * B_ + b) * I_ + o];
            __syncthreads();
        }

        float acc = bias;
        if (use_xg) {
            acc += xg[((size_t)t * B_ + b) * G_ + o];
            if (t + 1 < T_)
                __builtin_prefetch(&xg[((size_t)(t + 1) * B_ + b) * G_ + o], 0, 1);
        }

#pragma unroll 8
        for (int k = 0; k < H_; ++k)
            acc += h_lds[k] * wh[(size_t)k * G_];

        if (!use_xg) {
#pragma unroll 8
            for (int k = 0; k < I_; ++k)
                acc += x_lds[k] * wx[(size_t)k * G_];
        }

        gates[o] = acc;
        __syncthreads();   // all dot-products done; h_lds safe to overwrite

        if (o < H_) {
            float ig = 1.f / (1.f + __expf(-gates[o]));
            float fg = 1.f / (1.f + __expf(-gates[H_ + o]));
            float gg = tanhf(gates[2 * H_ + o]);
            float og = 1.f / (1.f + __expf(-gates[3 * H_ + o]));
            c = fg * c + ig * gg;
            float h = og * tanhf(c);
            h_lds[o] = h;
            out[((size_t)t * B_ + b) * H_ + o] = h;
        }
        __syncthreads();
    }

    if (o < H_) {
        const size_t TBH = (size_t)T_ * B_ * H_;
        out[TBH + (size_t)b * H_ + o] = h_lds[o];           // final h
        out[TBH + (size_t)B_ * H_ + (size_t)b * H_ + o] = c; // final c
    }
}

// ---------------------------------------------------------------------------
extern "C" void kernel_launch(void* const* d_in, const int* in_sizes, int n_in,
                              void* d_out, int out_size, void* d_ws, size_t ws_size,
                              hipStream_t stream) {
    const float* x    = (const float*)d_in[0];
    const float* h0   = (const float*)d_in[1];
    const float* c0   = (const float*)d_in[2];
    const float* coef = (const float*)d_in[3];
    const float* Wx   = (const float*)d_in[4];
    const float* bx   = (const float*)d_in[5];
    const float* Wh   = (const float*)d_in[6];
    const float* bh   = (const float*)d_in[7];
    float* out = (float*)d_out;
    float* ws  = (float*)d_ws;

    const size_t W1      = (size_t)B_ * I_ * G_;   // 8,388,608 floats per matrix
    const size_t xgElems = (size_t)T_ * B_ * G_;   // 67,108,864 floats

    // Deterministic (ws_size is fixed per harness instance).
    int use_xg = (ws_size >= (2 * W1 + xgElems) * sizeof(float)) ? 1 : 0;

    float* WxT = ws;
    float* WhT = ws + W1;
    float* xg  = ws + 2 * W1;

    // Kernel 1: 2 * B*I*G elements = 16,777,216 -> 65536 blocks of 256
    mow_eff_weights<<<65536, 256, 0, stream>>>(coef, Wx, Wh, WxT, WhT);

    // Kernel 2: 65536 waves (32 batches x 128 m-tiles x 16 n-strips), 8 waves/block
    if (use_xg)
        mow_xproj_gemm<<<8192, 256, 0, stream>>>(x, WxT, xg);

    // Kernel 3: one workgroup per batch
    mow_recurrence<<<B_, 1024, 0, stream>>>(x, xg, h0, c0, coef, bx, bh,
                                            WxT, WhT, out, use_xg);
}